// LinkPrediction_73289321939192
// MI455X (gfx1250) — compile-verified
//
#include <hip/hip_runtime.h>
#include <climits>

// ---------------------------------------------------------------------------
// R-GCN link prediction for MI455X (gfx1250, wave32).
// fp32 end-to-end; matrix work on V_WMMA_F32_16X16X4_F32.
// Each WMMA wave computes a 16x64 output strip (4 accumulators) so A
// fragments are loaded once per 4 WMMAs -> 4x less L2 gather traffic.
// h table (51.2MB) is L2-resident on the 192MB L2, so edge gathers/scatters
// never touch HBM in steady state.
// ---------------------------------------------------------------------------

typedef __attribute__((ext_vector_type(2))) float v2f;
typedef __attribute__((ext_vector_type(8))) float v8f;

#define DD 256   // hidden dim
#define RR 8     // num relations
#define CT 4     // col tiles per wave (16x64 strip)

__device__ __forceinline__ v8f wmma_f32_k4(v2f a, v2f b, v8f c) {
  // 8-arg pattern: (neg_a, A, neg_b, B, c_mod, C, reuse_a, reuse_b)
  return __builtin_amdgcn_wmma_f32_16x16x4_f32(false, a, false, b, (short)0, c,
                                               false, false);
}

// ---- graph preprocessing ---------------------------------------------------

__global__ void k_init(float* counts, int* relCnt, int* cursor, int* firstOcc,
                       int total) {
  int i = blockIdx.x * blockDim.x + threadIdx.x;
  if (i < total) counts[i] = 0.0f;
  if (i < RR) { relCnt[i] = 0; cursor[i] = 0; firstOcc[i] = INT_MAX; }
}

__global__ void k_count(const int* __restrict__ dst, const int* __restrict__ et,
                        int E, int N, float* counts, int* relCnt,
                        int* firstOcc) {
  int e = blockIdx.x * blockDim.x + threadIdx.x;
  if (e >= E) return;
  int r = et[e];
  atomicAdd(&counts[(size_t)r * N + dst[e]], 1.0f);
  atomicAdd(&relCnt[r], 1);
  atomicMin(&firstOcc[r], e);
}

__global__ void k_scan(const int* relCnt, int* relOff, int* tileOff,
                       int* totalTiles) {
  relOff[0] = 0; tileOff[0] = 0;
  for (int r = 0; r < RR; r++) {
    relOff[r + 1]  = relOff[r] + relCnt[r];
    tileOff[r + 1] = tileOff[r] + (relCnt[r] + 15) / 16;
  }
  *totalTiles = tileOff[RR];
}

__global__ void k_sort(const int* __restrict__ et, int E,
                       const int* __restrict__ relOff, int* cursor,
                       int* sorted) {
  int e = blockIdx.x * blockDim.x + threadIdx.x;
  if (e >= E) return;
  int r = et[e];
  int p = atomicAdd(&cursor[r], 1);
  sorted[relOff[r] + p] = e;
}

// Faithful to reference overwrite semantics: among relations present at node v,
// the one with the LARGEST first-occurrence edge index wins (jnp.argmax ties ->
// first index, so strict '>' with bestKey init below matches).
__global__ void k_norm(const float* __restrict__ counts,
                       const int* __restrict__ firstOcc, int N,
                       float* __restrict__ norm) {
  int v = blockIdx.x * blockDim.x + threadIdx.x;
  if (v >= N) return;
  int best = 0; float bestKey = -2.0f;
  for (int r = 0; r < RR; r++) {
    float c = counts[(size_t)r * N + v];
    float key = (c > 0.0f) ? (float)firstOcc[r] : -1.0f;
    if (key > bestKey) { bestKey = key; best = r; }
  }
  float c = counts[(size_t)best * N + v];
  norm[v] = (c > 0.0f) ? (1.0f / c) : 0.0f;
}

// ---- dense GEMM: C[M x 256] = A[M x 256] @ B[256 x 256] (+bias) ------------
// One wave per 16x64 output strip; K accumulated in 64 WMMA f32 16x16x4 ops
// per 16x16 tile, A fragment reused across the 4 tiles of the strip.
// A fragment (ISA 16x4 f32): lane half -> K pair, VGPR -> K within pair,
// lane%16 -> M. B fragment symmetric with lane%16 -> N.
// C: VGPR i, lanes0-15 -> M=i, lanes16-31 -> M=i+8, N=lane%16.
__global__ void k_gemm(const float* __restrict__ A, const float* __restrict__ Bm,
                       const float* __restrict__ bias, float* __restrict__ C) {
  const int l    = threadIdx.x;      // 0..31
  const int n    = l & 15;
  const int half = l >> 4;
  const int klo  = half * 2;
  const int rowBase = blockIdx.x * 16;
  const int cb      = blockIdx.y * (16 * CT);
  const float* Arow = A + (size_t)(rowBase + n) * DD;
  const float* Bc   = Bm + cb + n;
  v8f acc[CT];
#pragma unroll
  for (int t = 0; t < CT; t++) acc[t] = (v8f){};
#pragma unroll 2
  for (int k = 0; k < DD; k += 4) {
    v2f a; a.x = Arow[k + klo]; a.y = Arow[k + klo + 1];
    const float* B0 = Bc + (size_t)(k + klo) * DD;
    const float* B1 = Bc + (size_t)(k + klo + 1) * DD;
#pragma unroll
    for (int t = 0; t < CT; t++) {
      v2f b = {B0[t * 16], B1[t * 16]};
      acc[t] = wmma_f32_k4(a, b, acc[t]);
    }
  }
#pragma unroll
  for (int t = 0; t < CT; t++) {
    const int col = cb + t * 16 + n;
    const float bv = bias ? bias[col] : 0.0f;
#pragma unroll
    for (int i = 0; i < 8; i++) {
      int row = rowBase + i + half * 8;
      C[(size_t)row * DD + col] = acc[t][i] + bv;
    }
  }
}

// ---- relation message transform: WMMA over relation-sorted edge tiles ------
// hnext[dst] += norm[dst] * (h[src] @ W[rel]) for every edge; 16 edges/tile,
// each wave produces that tile's 16x64 output strip.
__global__ void k_edge(const float* __restrict__ h, const float* __restrict__ W,
                       const float* __restrict__ norm,
                       const int* __restrict__ src, const int* __restrict__ dst,
                       const int* __restrict__ sorted,
                       const int* __restrict__ relCnt,
                       const int* __restrict__ relOff,
                       const int* __restrict__ tileOff,
                       const int* __restrict__ totalTiles,
                       float* __restrict__ hnext) {
  const int t = blockIdx.x;
  if (t >= *totalTiles) return;                 // wave-uniform exit
  int r = 0;
  while (r < RR - 1 && t >= tileOff[r + 1]) r++;
  const int lt   = t - tileOff[r];
  const int cnt  = relCnt[r];
  const int base = relOff[r];

  const int l    = threadIdx.x;
  const int n    = l & 15;
  const int half = l >> 4;
  const int klo  = half * 2;
  const int cb   = blockIdx.y * (16 * CT);

  // This lane's A-matrix row (edge lt*16 + n). Padding rows read node 0's
  // features; their C rows are skipped at scatter, so values are harmless and
  // EXEC stays all-ones through the WMMA loop.
  const int srow  = lt * 16 + n;
  const int eA    = (srow < cnt) ? sorted[base + srow] : 0;
  const int nodeA = (srow < cnt) ? src[eA] : 0;
  const float* Arow = h + (size_t)nodeA * DD;
  const float* Wc   = W + (size_t)r * DD * DD + cb + n;

  v8f acc[CT];
#pragma unroll
  for (int c = 0; c < CT; c++) acc[c] = (v8f){};
#pragma unroll 2
  for (int k = 0; k < DD; k += 4) {
    v2f a; a.x = Arow[k + klo]; a.y = Arow[k + klo + 1];
    const float* B0 = Wc + (size_t)(k + klo) * DD;
    const float* B1 = Wc + (size_t)(k + klo + 1) * DD;
#pragma unroll
    for (int c = 0; c < CT; c++) {
      v2f b = {B0[c * 16], B1[c * 16]};
      acc[c] = wmma_f32_k4(a, b, acc[c]);
    }
  }
#pragma unroll
  for (int i = 0; i < 8; i++) {
    int row = lt * 16 + i + half * 8;
    if (row < cnt) {
      int e  = sorted[base + row];
      int dv = dst[e];
      float nm = norm[dv];
      float* outRow = hnext + (size_t)dv * DD + cb + n;
#pragma unroll
      for (int c = 0; c < CT; c++)
        atomicAdd(outRow + c * 16, acc[c][i] * nm);
    }
  }
}

// ---- basis combine: W[r] = sum_b coe[r,b] * bases[b] -----------------------
__global__ void k_makeW(const float* __restrict__ bases,
                        const float* __restrict__ coe, int Bn,
                        float* __restrict__ W) {
  size_t i = (size_t)blockIdx.x * blockDim.x + threadIdx.x;
  if (i >= (size_t)RR * DD * DD) return;
  int r = (int)(i / ((size_t)DD * DD));
  size_t j = i % ((size_t)DD * DD);
  float s = 0.0f;
  for (int b = 0; b < Bn; b++)
    s += coe[r * Bn + b] * bases[(size_t)b * DD * DD + j];
  W[i] = s;
}

// ---- batchnorm (inference) + relu ------------------------------------------
__global__ void k_bnrelu(const float* __restrict__ x, const float* __restrict__ g,
                         const float* __restrict__ bta,
                         const float* __restrict__ mean,
                         const float* __restrict__ var,
                         float* __restrict__ out, size_t total) {
  size_t i = (size_t)blockIdx.x * blockDim.x + threadIdx.x;
  if (i >= total) return;
  int c = (int)(i % DD);
  float v = (x[i] - mean[c]) * rsqrtf(var[c] + 1e-3f) * g[c] + bta[c];
  out[i] = v > 0.0f ? v : 0.0f;
}

// ---- DistMult scoring: one wave32 per triple -------------------------------
__global__ void k_score(const float* __restrict__ h,
                        const float* __restrict__ wrel,
                        const int* __restrict__ src,
                        const int* __restrict__ other,
                        const int* __restrict__ et, int count, int K,
                        float* __restrict__ out) {
  int wid  = blockIdx.x * (blockDim.x >> 5) + (threadIdx.x >> 5);
  int lane = threadIdx.x & 31;
  if (wid >= count) return;                    // wave-uniform
  int e = (K > 1) ? (wid / K) : wid;
  const float* hs = h + (size_t)src[e] * DD;
  const float* hd = h + (size_t)other[wid] * DD;
  const float* wr = wrel + (size_t)et[e] * DD;
  float s = 0.0f;
#pragma unroll
  for (int d = lane; d < DD; d += 32) s += hs[d] * hd[d] * wr[d];
  for (int off = 16; off; off >>= 1) s += __shfl_xor(s, off, 32);
  if (lane == 0) out[wid] = s;
}

// ---------------------------------------------------------------------------

extern "C" void kernel_launch(void* const* d_in, const int* in_sizes, int n_in,
                              void* d_out, int out_size, void* d_ws,
                              size_t ws_size, hipStream_t stream) {
  (void)n_in; (void)out_size; (void)ws_size;
  const float* node_feat  = (const float*)d_in[0];
  const float* w_emb      = (const float*)d_in[1];
  const float* b_emb      = (const float*)d_in[2];
  const float* w_bases    = (const float*)d_in[3];
  const float* w_coe      = (const float*)d_in[4];
  const float* w_self     = (const float*)d_in[5];
  const float* bn_gamma   = (const float*)d_in[6];
  const float* bn_beta    = (const float*)d_in[7];
  const float* bn_mean    = (const float*)d_in[8];
  const float* bn_var     = (const float*)d_in[9];
  const float* w_relation = (const float*)d_in[10];
  const int*   src        = (const int*)d_in[11];
  const int*   dst        = (const int*)d_in[12];
  const int*   et         = (const int*)d_in[13];
  const int*   neg_dst    = (const int*)d_in[14];

  const int D  = DD;
  const int N  = in_sizes[0] / D;
  const int E  = in_sizes[11];
  const int L  = in_sizes[5] / (D * D);
  const int Bn = in_sizes[4] / (L * RR);
  const int K  = in_sizes[14] / E;
  float* out = (float*)d_out;

  // workspace carve-up (~108 MB total for the reference sizes)
  float* h      = (float*)d_ws;
  float* hnext  = h + (size_t)N * D;
  float* W      = hnext + (size_t)N * D;
  float* counts = W + (size_t)RR * D * D;
  float* norm   = counts + (size_t)RR * N;
  int* sorted     = (int*)(norm + N);
  int* relCnt     = sorted + E;
  int* relOff     = relCnt + RR;          // RR+1 entries
  int* tileOff    = relOff + RR + 1;      // RR+1 entries
  int* cursor     = tileOff + RR + 1;
  int* firstOcc   = cursor + RR;
  int* totalTiles = firstOcc + RR;

  const int rn = RR * N;
  k_init<<<(rn + 255) / 256, 256, 0, stream>>>(counts, relCnt, cursor, firstOcc, rn);
  k_count<<<(E + 255) / 256, 256, 0, stream>>>(dst, et, E, N, counts, relCnt, firstOcc);
  k_scan<<<1, 1, 0, stream>>>(relCnt, relOff, tileOff, totalTiles);
  k_sort<<<(E + 255) / 256, 256, 0, stream>>>(et, E, relOff, cursor, sorted);
  k_norm<<<(N + 255) / 256, 256, 0, stream>>>(counts, firstOcc, N, norm);

  dim3 gEmb(N / 16, D / (16 * CT));
  k_gemm<<<gEmb, 32, 0, stream>>>(node_feat, w_emb, b_emb, h);

  const int maxTiles = (E + 15) / 16 + RR;    // upper bound incl. per-rel pad
  const size_t nd = (size_t)N * D;
  for (int l = 0; l < L; l++) {
    size_t rdd = (size_t)RR * D * D;
    k_makeW<<<(unsigned)((rdd + 255) / 256), 256, 0, stream>>>(
        w_bases + (size_t)l * Bn * D * D, w_coe + (size_t)l * RR * Bn, Bn, W);
    k_gemm<<<gEmb, 32, 0, stream>>>(h, w_self + (size_t)l * D * D, nullptr, hnext);
    dim3 gE(maxTiles, D / (16 * CT));
    k_edge<<<gE, 32, 0, stream>>>(h, W, norm, src, dst, sorted, relCnt, relOff,
                                  tileOff, totalTiles, hnext);
    k_bnrelu<<<(unsigned)((nd + 255) / 256), 256, 0, stream>>>(
        hnext, bn_gamma + l * D, bn_beta + l * D, bn_mean + l * D,
        bn_var + l * D, h, nd);
  }

  k_score<<<(E + 7) / 8, 256, 0, stream>>>(h, w_relation, src, dst, et, E, 1, out);
  k_score<<<(E * K + 7) / 8, 256, 0, stream>>>(h, w_relation, src, neg_dst, et,
                                               E * K, K, out + E);
}